// CosmosPatcher3d_13056700579891
// MI455X (gfx1250) — compile-verified
//
#include <hip/hip_runtime.h>

typedef float v2f __attribute__((ext_vector_type(2)));
typedef float v8f __attribute__((ext_vector_type(8)));

// Input : (1, 3, 33, 704, 704) f32
// Output: (1, 192, 9, 176, 176) f32
//
// Two-level 3D Haar DWT == disjoint 4x4x4 block transform:
//   out(ch) = (1/64) * sum_e sign(ch,e) * x(e),  sign(ch,e) = (-1)^popc(ch & e)
// ch = tq*16 + hq*4 + wq, e = i*16 + j*4 + k. Output plane = s2*24 + s1*3 + c
// (s2/s1 = bit-deinterleave of ch). T prefix pad: t_src = max(4*bt + i - 3, 0).
//
// Per wave: Out[64ch x 16blk] = M64 * X[64e x 16blk] via v_wmma_f32_16x16x4_f32.
// A(ks,ct) = (-1)^popc(ct&i) * a[j]  -> only 8 distinct A fragments, precomputed.
// B tile per K-step = one fully coalesced 256B wave load already in B layout.

__global__ __launch_bounds__(256) void haar3d_dwt2_wmma(
    const float* __restrict__ in, float* __restrict__ out) {
  const int lane = threadIdx.x & 31;
  const int half = lane >> 4;     // 0: lanes 0-15 (K=0,1), 1: lanes 16-31 (K=2,3)
  const int m    = lane & 15;     // A row (channel) / B and D column (block)

  // Wave-uniform tile id forced into SGPRs -> all address bases become SALU.
  const int wid  = __builtin_amdgcn_readfirstlane((int)(threadIdx.x >> 5));
  int tile = (int)blockIdx.x * 8 + wid;
  const int tw = tile % 11;  tile /= 11;            // 176/16 = 11 W-tiles
  const int bh = tile % 176; tile /= 176;
  const int bt = tile % 9;
  const int c  = tile / 9;
  const int bw0 = tw * 16;

  // Per-lane input offset within a row (floats): block m, elements k=2h,2h+1
  const int woff = 4 * (bw0 + m) + 2 * half;

  // Precompute the 8 possible A fragments: +/- a[j], j = h-offset in block.
  constexpr float S = 1.0f / 64.0f;
  v2f aj[4], an[4];
#pragma unroll
  for (int j = 0; j < 4; ++j) {
    const int mask0 = 4 * j + 2 * half;        // e&15 of lane's first element
    const float s0 = (__popc(m & mask0) & 1) ? -S : S;
    const float s1 = (__popc(m & (mask0 + 1)) & 1) ? -S : S;
    aj[j].x = s0;  aj[j].y = s1;
    an[j].x = -s0; an[j].y = -s1;
  }

  v8f acc[4] = {};   // D tiles: channels [16ct..16ct+15] x 16 blocks

#pragma unroll
  for (int ks = 0; ks < 16; ++ks) {
    const int i = ks >> 2;         // t offset within block
    const int j = ks & 3;          // h offset within block
    int tg = 4 * bt + i - 3; if (tg < 0) tg = 0;   // T source with repeat-pad
    const float* row =
        in + (((size_t)c * 33 + tg) * 704 + (size_t)(4 * bh + j)) * 704;
    const v2f b = *(const v2f*)(row + woff);       // coalesced 256B wave load

    if (i == 0) {
      // popc(ct & 0) == 0 for every ct: all 4 accumulators get the same
      // contribution -> accumulate once, fork after the i==0 group.
      acc[0] = __builtin_amdgcn_wmma_f32_16x16x4_f32(
          false, aj[j], false, b, (short)0, acc[0], false, false);
    } else {
#pragma unroll
      for (int ct = 0; ct < 4; ++ct) {
        const bool neg = (__popc(ct & i) & 1) != 0;   // compile-time constant
        acc[ct] = __builtin_amdgcn_wmma_f32_16x16x4_f32(
            false, neg ? an[j] : aj[j], false, b, (short)0, acc[ct],
            false, false);
      }
    }
    if (ks == 3) { acc[1] = acc[0]; acc[2] = acc[0]; acc[3] = acc[0]; }
  }

  // Store. VGPR v of acc[ct]: channel 16ct+v (lanes 0-15) / +8 (lanes 16-31),
  // block column m. plane(ch+8) - plane(ch) == 48 always, so the half
  // dependence folds into one per-lane offset; plane0 is a scalar constant.
  const size_t SP = (size_t)9 * 176 * 176;                 // plane stride
  const size_t lane_off = (size_t)(bw0 + m) + (size_t)(48 * half) * SP;
  float* obase = out + (size_t)bt * (176 * 176) + (size_t)bh * 176;
#pragma unroll
  for (int ct = 0; ct < 4; ++ct) {
#pragma unroll
    for (int v = 0; v < 8; ++v) {
      const int ch0 = 16 * ct + v;                         // half==0 channel
      const int s2 = ((ch0 >> 5) & 1) * 4 + ((ch0 >> 3) & 1) * 2 + ((ch0 >> 1) & 1);
      const int s1 = ((ch0 >> 4) & 1) * 4 + ((ch0 >> 2) & 1) * 2 + (ch0 & 1);
      const int plane0 = s2 * 24 + s1 * 3 + c;             // scalar
      obase[(size_t)plane0 * SP + lane_off] = acc[ct][v];
    }
  }
}

extern "C" void kernel_launch(void* const* d_in, const int* in_sizes, int n_in,
                              void* d_out, int out_size, void* d_ws, size_t ws_size,
                              hipStream_t stream) {
  (void)in_sizes; (void)n_in; (void)d_ws; (void)ws_size; (void)out_size;
  const float* in = (const float*)d_in[0];
  float* out = (float*)d_out;
  // tiles = 3 * 9 * 176 * 11 = 52272 waves; 8 waves per 256-thread block
  haar3d_dwt2_wmma<<<dim3(52272 / 8), dim3(256), 0, stream>>>(in, out);
}